// NeuroselectiveLinear_21663815041244
// MI455X (gfx1250) — compile-verified
//
#include <hip/hip_runtime.h>
#include <stdint.h>

typedef __attribute__((ext_vector_type(2))) float v2f;
typedef __attribute__((ext_vector_type(4))) float v4f;
typedef __attribute__((ext_vector_type(8))) float v8f;
typedef __attribute__((ext_vector_type(4))) int   v4i;

#define IN_F    4096
#define OUT_F   4096
#define ACT_IN  1024
#define ACT_OUT 1024
#define M_TOTAL 8192   // BATCH * SEQ = 4 * 2048

// Block tile: 128 (M) x 128 (N), K-tile 16. 8 waves -> 4 (M) x 2 (N), each wave 32x64.
#define BM 128
#define BN 128
#define BK 16
#define NTILES (ACT_IN / BK)
#define LDSD 20        // padded LDS row stride: rows 80B apart -> 16B-aligned, bank-conflict-free

// ---- CDNA5 async global->LDS path (ASYNCcnt), with sync fallback --------------------
#if defined(__has_builtin)
#  if __has_builtin(__builtin_amdgcn_global_load_async_to_lds_b128) && \
      __has_builtin(__builtin_amdgcn_global_load_async_to_lds_b32)
#    define HAVE_ASYNC_LDS 1
#  endif
#  if __has_builtin(__builtin_amdgcn_s_wait_asynccnt)
#    define HAVE_WAIT_ASYNC_BUILTIN 1
#  endif
#endif
#ifndef HAVE_ASYNC_LDS
#  define HAVE_ASYNC_LDS 0
#endif
#ifndef HAVE_WAIT_ASYNC_BUILTIN
#  define HAVE_WAIT_ASYNC_BUILTIN 0
#endif

// address-space-qualified typed pointers matching the builtin signatures
typedef __attribute__((address_space(1))) v4i* g_v4i_p;   // global int4*
typedef __attribute__((address_space(3))) v4i* l_v4i_p;   // LDS    int4*
typedef __attribute__((address_space(1))) int* g_i32_p;   // global int*
typedef __attribute__((address_space(3))) int* l_i32_p;   // LDS    int*

__device__ __forceinline__ void async_copy_b128(const void* g, void* l)
{
#if HAVE_ASYNC_LDS
    __builtin_amdgcn_global_load_async_to_lds_b128(
        (g_v4i_p)(void*)g, (l_v4i_p)l, 0, 0);
#else
    *(v4f*)l = *(const v4f*)g;
#endif
}

__device__ __forceinline__ void async_copy_b32(const void* g, void* l)
{
#if HAVE_ASYNC_LDS
    __builtin_amdgcn_global_load_async_to_lds_b32(
        (g_i32_p)(void*)g, (l_i32_p)l, 0, 0);
#else
    *(float*)l = *(const float*)g;
#endif
}

__device__ __forceinline__ void wait_async0()
{
#if HAVE_ASYNC_LDS
#  if HAVE_WAIT_ASYNC_BUILTIN
    __builtin_amdgcn_s_wait_asynccnt(0);
#  else
    asm volatile("s_wait_asynccnt 0" ::: "memory");
#  endif
#endif
}

// ---------------------------------------------------------------------------
// Kernel 1: zero-fill the full [M, 4096] output (write-once -> non-temporal)
// ---------------------------------------------------------------------------
__global__ __launch_bounds__(256) void zero_out_kernel(float* __restrict__ out, size_t n)
{
    size_t i = ((size_t)blockIdx.x * 256 + threadIdx.x) * 4;
    if (i < n) {
        v4f z = {0.f, 0.f, 0.f, 0.f};
        __builtin_nontemporal_store(z, (v4f*)(out + i));
    }
}

// ---------------------------------------------------------------------------
// Kernel 2: pre-gather active input columns: Xg[m][k] = x[m][in_idx[k]]
// ---------------------------------------------------------------------------
__global__ __launch_bounds__(256) void gather_x_kernel(const float* __restrict__ x,
                                                       const int*   __restrict__ in_idx,
                                                       float*       __restrict__ xg)
{
    int k = blockIdx.x * 256 + threadIdx.x;   // 0..ACT_IN-1 (grid.x = ACT_IN/256)
    int m = blockIdx.y;                       // 0..M_TOTAL-1
    xg[(size_t)m * ACT_IN + k] = x[(size_t)m * IN_F + in_idx[k]];
}

// ---------------------------------------------------------------------------
// Kernel 3: fp32 WMMA GEMM  C = A * W^T + b, scatter-stored through out_idx.
// Double-buffered LDS tiles filled by async global->LDS DMA.
// ---------------------------------------------------------------------------
template <bool GATHER>
__global__ __launch_bounds__(256) void nslinear_wmma_kernel(
    const float* __restrict__ A, int lda,
    const int*   __restrict__ in_idx,   // used only when GATHER
    const float* __restrict__ W,
    const float* __restrict__ bias,
    const int*   __restrict__ out_idx,
    float*       __restrict__ out)
{
    __shared__ __align__(16) float As[2][BM][LDSD];   // [buf][m][k]
    __shared__ __align__(16) float Bs[2][BN][LDSD];   // [buf][n][k]

    const int tid   = threadIdx.x;
    const int wave  = tid >> 5;
    const int lane  = tid & 31;
    const int lo    = lane & 15;       // fragment row/col within 16
    const int hi    = lane >> 4;       // K-pair select (K={0,1} vs K={2,3})
    const int waveM = wave >> 1;       // 0..3
    const int waveN = wave & 1;        // 0..1
    const int m0    = blockIdx.y * BM;
    const int n0    = blockIdx.x * BN;

    v8f acc[2][4];
#pragma unroll
    for (int mi = 0; mi < 2; ++mi)
#pragma unroll
        for (int ni = 0; ni < 4; ++ni)
            acc[mi][ni] = (v8f){0.f, 0.f, 0.f, 0.f, 0.f, 0.f, 0.f, 0.f};

    // issue all async copies for one K-tile into LDS buffer `buf`
    auto stage = [&](int buf, int kbase) {
        if (GATHER) {
            const int k   = tid & 15;
            const int r0  = tid >> 4;
            const int col = in_idx[kbase + k];
#pragma unroll
            for (int i = 0; i < 8; ++i) {
                const int r = r0 + i * 16;
                async_copy_b32(&A[(size_t)(m0 + r) * lda + col], &As[buf][r][k]);
            }
        } else {
#pragma unroll
            for (int i = 0; i < 2; ++i) {
                const int idx2 = tid + i * 256;     // 0..511
                const int r    = idx2 >> 2;         // 0..127
                const int kg   = (idx2 & 3) * 4;    // 0,4,8,12
                async_copy_b128(&A[(size_t)(m0 + r) * lda + kbase + kg], &As[buf][r][kg]);
            }
        }
#pragma unroll
        for (int i = 0; i < 2; ++i) {
            const int idx2 = tid + i * 256;
            const int n    = idx2 >> 2;
            const int kg   = (idx2 & 3) * 4;
            async_copy_b128(&W[(size_t)(n0 + n) * ACT_IN + kbase + kg], &Bs[buf][n][kg]);
        }
    };

    stage(0, 0);   // prologue: fill buffer 0

    for (int kt = 0; kt < NTILES; ++kt) {
        const int cur = kt & 1;

        wait_async0();      // this wave's DMA into `cur` complete
        __syncthreads();    // everyone's DMA done + everyone finished reading `nxt`

        if (kt + 1 < NTILES)
            stage(cur ^ 1, (kt + 1) * BK);   // overlap next tile's DMA with compute

        // --- 4 WMMA K-steps of 4 over this K-tile ---
#pragma unroll
        for (int kk = 0; kk < BK; kk += 4) {
            v2f a[2], b[4];
#pragma unroll
            for (int mi = 0; mi < 2; ++mi) {
                const int r = waveM * 32 + mi * 16 + lo;
                a[mi] = *(const v2f*)&As[cur][r][kk + 2 * hi];
            }
#pragma unroll
            for (int ni = 0; ni < 4; ++ni) {
                const int c = waveN * 64 + ni * 16 + lo;
                b[ni] = *(const v2f*)&Bs[cur][c][kk + 2 * hi];
            }
#pragma unroll
            for (int mi = 0; mi < 2; ++mi)
#pragma unroll
                for (int ni = 0; ni < 4; ++ni)
                    acc[mi][ni] = __builtin_amdgcn_wmma_f32_16x16x4_f32(
                        false, a[mi], false, b[ni],
                        (short)0, acc[mi][ni], false, false);
        }
    }

    // --- epilogue: add bias, scatter through out_idx (non-temporal, write-once) ---
#pragma unroll
    for (int ni = 0; ni < 4; ++ni) {
        const int   nact = n0 + waveN * 64 + ni * 16 + lo;   // active-output index
        const float bv   = bias[nact];
        const int   ocol = out_idx[nact];
#pragma unroll
        for (int mi = 0; mi < 2; ++mi) {
            const int mbase = m0 + waveM * 32 + mi * 16 + 8 * hi;
#pragma unroll
            for (int r = 0; r < 8; ++r)
                __builtin_nontemporal_store(acc[mi][ni][r] + bv,
                                            &out[(size_t)(mbase + r) * OUT_F + ocol]);
        }
    }
}

// ---------------------------------------------------------------------------
extern "C" void kernel_launch(void* const* d_in, const int* in_sizes, int n_in,
                              void* d_out, int out_size, void* d_ws, size_t ws_size,
                              hipStream_t stream)
{
    const float* x       = (const float*)d_in[0];   // [4, 2048, 4096]
    const float* W       = (const float*)d_in[1];   // [1024, 1024]
    const float* bias    = (const float*)d_in[2];   // [1024]
    const int*   in_idx  = (const int*)d_in[3];     // [1024]
    const int*   out_idx = (const int*)d_in[4];     // [1024]
    float*       out     = (float*)d_out;           // [4, 2048, 4096]

    (void)in_sizes; (void)n_in; (void)out_size;

    // 1) zero the output buffer
    const size_t outElems = (size_t)M_TOTAL * OUT_F;
    const int zeroBlocks = (int)((outElems / 4 + 255) / 256);
    zero_out_kernel<<<zeroBlocks, 256, 0, stream>>>(out, outElems);

    // 2) pre-gather active columns into workspace if it fits (32 MiB)
    const size_t xgBytes = (size_t)M_TOTAL * ACT_IN * sizeof(float);
    const dim3 gemmGrid(ACT_OUT / BN, M_TOTAL / BM);

    if (ws_size >= xgBytes) {
        float* xg = (float*)d_ws;
        gather_x_kernel<<<dim3(ACT_IN / 256, M_TOTAL), 256, 0, stream>>>(x, in_idx, xg);
        nslinear_wmma_kernel<false><<<gemmGrid, 256, 0, stream>>>(
            xg, ACT_IN, nullptr, W, bias, out_idx, out);
    } else {
        // gather inside the GEMM (per-element async b32 DMA through in_idx)
        nslinear_wmma_kernel<true><<<gemmGrid, 256, 0, stream>>>(
            x, IN_F, in_idx, W, bias, out_idx, out);
    }
}